// Deployable_Network_71992241815954
// MI455X (gfx1250) — compile-verified
//
#include <hip/hip_runtime.h>
#include <stdint.h>

#define N_BOXES 8192
#define WORDS   256      // N/32 mask words per row
#define TILE    2048     // boxes staged per LDS tile (40KB: 32KB boxes + 8KB labels)
#define NTILES  4
#define SCORE_THRESH 0.5f
#define IOU_THRESH   0.5f

// ---------- CDNA5 helpers ----------

static __device__ __forceinline__ uint32_t ballot32(bool p) {
#if defined(__has_builtin) && __has_builtin(__builtin_amdgcn_ballot_w32)
  return __builtin_amdgcn_ballot_w32(p);
#else
  return (uint32_t)__ballot(p);
#endif
}

// broadcast lane srcLane's 32-bit value to all lanes (wave32)
static __device__ __forceinline__ uint32_t bcast32(uint32_t v, int srcLane) {
  return (uint32_t)__builtin_amdgcn_ds_bpermute(srcLane << 2, (int)v);
}

// gfx1250 async global->LDS copy (ASYNCcnt path); 16B per lane.
// VDST = LDS byte offset VGPR, VADDR = 64-bit global address VGPR pair (GV mode).
static __device__ __forceinline__ void async_copy_b128(void* lds_dst, const void* gsrc) {
  uint32_t loff = (uint32_t)(uintptr_t)lds_dst;     // low 32 bits of generic LDS ptr = LDS offset
  uint64_t ga   = (uint64_t)(uintptr_t)gsrc;
  asm volatile("global_load_async_to_lds_b128 %0, %1, off"
               : : "v"(loff), "v"(ga) : "memory");
}

static __device__ __forceinline__ void wait_async0() {
  asm volatile("s_wait_asynccnt 0x0" : : : "memory");
}

// ---------- Phase 1: suppression bitmask build ----------
// mask[i][w] bit b == 1  iff  box i suppresses box j=w*32+b  (j>i, same class, IoU>thresh).
// Grid: 256 blocks x 256 threads (8 waves). Block b owns rows [b*32, b*32+32); wave owns 4 rows.
__global__ __launch_bounds__(256) void nms_mask_kernel(
    const float4* __restrict__ boxes, const int* __restrict__ labels,
    uint32_t* __restrict__ mask) {
  __shared__ float4 sBox[TILE];
  __shared__ int    sLab[TILE];
  const int lane = threadIdx.x & 31;
  const int wave = threadIdx.x >> 5;
  const int i0   = blockIdx.x * 32 + wave * 4;

  for (int t = 0; t < NTILES; ++t) {
    __syncthreads();  // previous tile fully consumed before overwrite
    // stage boxes tile (TILE * 16B) via async-to-LDS
    for (int k = threadIdx.x; k < TILE; k += 256)
      async_copy_b128(&sBox[k], &boxes[t * TILE + k]);
    // stage labels tile (TILE * 4B) as 16B chunks
    {
      const float4* lab4  = (const float4*)(labels + t * TILE);
      float4*       sLab4 = (float4*)sLab;
      for (int k = threadIdx.x; k < TILE / 4; k += 256)
        async_copy_b128(&sLab4[k], &lab4[k]);
    }
    wait_async0();
    __syncthreads();

    for (int r = 0; r < 4; ++r) {
      const int    i     = i0 + r;
      const float4 bi    = boxes[i];       // L2-hot scalar-ish load
      const int    li    = labels[i];
      const float  areai = (bi.z - bi.x) * (bi.w - bi.y);
      uint32_t* rowp = mask + (size_t)i * WORDS + t * (TILE / 32);

      for (int wb = 0; wb < TILE / 32; wb += 32) {   // groups of 32 words
        // wave-uniform skip: whole group lies at j <= i -> all zero (upper triangle only)
        const int maxj = t * TILE + (wb + 31) * 32 + 31;
        if (maxj <= i) { rowp[wb + lane] = 0u; continue; }

        uint32_t myw = 0u;
        for (int w2 = 0; w2 < 32; ++w2) {
          const int    wl    = wb + w2;
          const int    jl    = wl * 32 + lane;       // column index within tile
          const int    j     = t * TILE + jl;
          const float4 bj    = sBox[jl];
          const float  areaj = (bj.z - bj.x) * (bj.w - bj.y);
          const float  xl = fmaxf(bi.x, bj.x), yl = fmaxf(bi.y, bj.y);
          const float  xr = fminf(bi.z, bj.z), yr = fminf(bi.w, bj.w);
          const float  iw = fmaxf(xr - xl, 0.0f), ih = fmaxf(yr - yl, 0.0f);
          const float  inter = iw * ih;
          const float  uni   = areai + areaj - inter;  // >= ~1, so no 1e-9 clamp needed
          const bool   pred  = (j > i) && (sLab[jl] == li) && (inter > IOU_THRESH * uni);
          const uint32_t word = ballot32(pred);        // one 32-bit mask word per ballot
          if (w2 == lane) myw = word;                  // word (wb+w2) lands in lane w2
        }
        rowp[wb + lane] = myw;                         // coalesced b32 store of 32 words
      }
    }
  }
}

// ---------- Phase 2: exact sequential scan, single wave, register-resident bitmap ----------
// Lane l of slot s owns mask word (s*32 + l). remv accumulates suppressed boxes.
__global__ void nms_scan_kernel(const float* __restrict__ scores,
                                const uint32_t* __restrict__ mask,
                                uint32_t* __restrict__ keepflag) {
  const int lane = threadIdx.x & 31;

  // valid-bit bitmap (scores > thresh) in registers, same word ownership as remv
  uint32_t vreg[8];
#pragma unroll
  for (int s = 0; s < 8; ++s) {
    uint32_t acc = 0u;
    for (int w2 = 0; w2 < 32; ++w2) {
      const int  w = s * 32 + w2;
      const bool p = scores[w * 32 + lane] > SCORE_THRESH;
      const uint32_t word = ballot32(p);
      if (w2 == lane) acc = word;
    }
    vreg[s] = acc;
  }

  uint32_t remv[8] = {0u, 0u, 0u, 0u, 0u, 0u, 0u, 0u};
#pragma unroll
  for (int s = 0; s < 8; ++s) {          // slot s covers boxes [s*1024, (s+1)*1024)
    for (int ii = 0; ii < 1024; ++ii) {
      const int i = s * 1024 + ii;
      const uint32_t* row = mask + (size_t)i * WORDS;
      __builtin_prefetch(row + 16 * WORDS, 0, 1);   // global_prefetch_b8, 16 rows ahead

      // row i only sets bits for j > i, i.e. words >= s*32 -> slots < s provably zero
      uint32_t rw[8];
#pragma unroll
      for (int s2 = s; s2 < 8; ++s2) rw[s2] = row[s2 * 32 + lane];

      const int      ol  = (i >> 5) & 31;           // lane owning word (i>>5) in slot s
      const uint32_t rsb = bcast32(remv[s], ol);
      const uint32_t vb  = bcast32(vreg[s], ol);
      const uint32_t bit = 1u << (i & 31);
      const bool     kept = ((rsb & bit) == 0u) && ((vb & bit) != 0u);
      const uint32_t km   = kept ? 0xFFFFFFFFu : 0u;
#pragma unroll
      for (int s2 = s; s2 < 8; ++s2) remv[s2] |= (rw[s2] & km);   // branchless OR

      if (lane == 0) keepflag[i] = kept ? 1u : 0u;
    }
  }
}

// ---------- Phase 3: write outputs: boxes*k (N*4) || scores*k (N) || keep (N) ----------
__global__ __launch_bounds__(256) void nms_write_kernel(
    const float4* __restrict__ boxes, const float* __restrict__ scores,
    const uint32_t* __restrict__ keepflag, float* __restrict__ out) {
  const int i = blockIdx.x * 256 + threadIdx.x;
  if (i >= N_BOXES) return;
  const float  k = keepflag[i] ? 1.0f : 0.0f;
  const float4 b = boxes[i];
  ((float4*)out)[i] = make_float4(b.x * k, b.y * k, b.z * k, b.w * k);
  out[4 * N_BOXES + i] = scores[i] * k;
  out[5 * N_BOXES + i] = k;
}

// ---------- host launcher ----------
extern "C" void kernel_launch(void* const* d_in, const int* in_sizes, int n_in,
                              void* d_out, int out_size, void* d_ws, size_t ws_size,
                              hipStream_t stream) {
  (void)in_sizes; (void)n_in; (void)out_size; (void)ws_size;
  const float4* boxes  = (const float4*)d_in[0];
  const float*  scores = (const float*)d_in[1];
  const int*    labels = (const int*)d_in[2];

  uint32_t* mask     = (uint32_t*)d_ws;                       // 8192*256*4 = 8 MB
  uint32_t* keepflag = mask + (size_t)N_BOXES * WORDS;        // + 32 KB

  nms_mask_kernel<<<N_BOXES / 32, 256, 0, stream>>>(boxes, labels, mask);
  nms_scan_kernel<<<1, 32, 0, stream>>>(scores, mask, keepflag);
  nms_write_kernel<<<N_BOXES / 256, 256, 0, stream>>>(boxes, scores, keepflag, (float*)d_out);
}